// HeteroNet_69561290326813
// MI455X (gfx1250) — compile-verified
//
#include <hip/hip_runtime.h>

typedef __attribute__((ext_vector_type(2))) float v2f;
typedef __attribute__((ext_vector_type(8))) float v8f;

#define LDSP 132   // LDS row pitch (floats): 128 + 4 pad -> conflict-free b-frag reads

// ---------------- TDM: DMA one 128x128 f32 W matrix into LDS with pitch-132 padding ----
#if defined(__has_builtin)
#if __has_builtin(__builtin_amdgcn_tensor_load_to_lds) && __has_builtin(__builtin_amdgcn_s_wait_tensorcnt)
#define USE_TDM 1
#endif
#endif
#ifndef USE_TDM
#define USE_TDM 0
#endif

#if USE_TDM
typedef __attribute__((ext_vector_type(4))) unsigned int u32x4;
typedef __attribute__((ext_vector_type(4))) int i32x4;
typedef __attribute__((ext_vector_type(8))) int i32x8;

__device__ __forceinline__ void tdm_load_w128(const float* gW, unsigned lds_byte_off) {
  unsigned long long ga = (unsigned long long)(uintptr_t)gW;
  u32x4 g0;
  g0[0] = 1u;                       // count=1 (valid user descriptor)
  g0[1] = lds_byte_off;             // D#.lds_addr (bytes)
  g0[2] = (unsigned)ga;             // global_addr[31:0]
  g0[3] = (unsigned)((ga >> 32) & 0x1FFFFFFull) | (2u << 30);  // addr[56:32] | type=2
  i32x8 g1;
  g1[0] = (2 << 16)                 // data_size = 4 bytes
        | (1 << 20)                 // pad_enable
        | (6 << 22)                 // pad_interval: 128 DWORDs
        | (3 << 25);                // pad_amount: 4 DWORDs  -> LDS pitch = 132 floats
  g1[1] = 128 << 16;                // tensor_dim0 = 128   (bits[63:48])
  g1[2] = 128 << 16;                // tensor_dim1 = 128   (bits[95:80])
  g1[3] = 128 << 16;                // tile_dim0  = 128    (bits[127:112])
  g1[4] = 128;                      // tile_dim1  = 128    (bits[143:128]); tile_dim2 = 0
  g1[5] = 128;                      // tensor_dim0_stride = 128
  g1[6] = 16384 << 16;              // tensor_dim1_stride (unused for 2D tile)
  g1[7] = 0;
  i32x4 z4 = {};
#if __clang_major__ >= 23
  i32x8 z8 = {};
  __builtin_amdgcn_tensor_load_to_lds(g0, g1, z4, z4, z8, 0);
#else
  __builtin_amdgcn_tensor_load_to_lds(g0, g1, z4, z4, 0);
#endif
}
#endif  // USE_TDM

// ---------------- elementwise ----------------
__global__ void leaky_kernel(const float* __restrict__ x, float* __restrict__ y, long n) {
  long i = (long)blockIdx.x * blockDim.x + threadIdx.x;
  if (i < n) { float v = x[i]; y[i] = v >= 0.0f ? v : 0.01f * v; }
}

__global__ void add2_kernel(const float* __restrict__ a, const float* __restrict__ b,
                            float* __restrict__ o, int n) {
  int i = blockIdx.x * blockDim.x + threadIdx.x;
  if (i < n) o[i] = a[i] + b[i];
}

// ---------------- effective-weight folding ----------------
// WnE = Wu[:, :128] @ Wn ; WsE = Wu[:, 128:] @ Ws ; bE = Wu1@bn + Wu2@bs + bu
__global__ void build_eff_kernel(const float* __restrict__ Wn, const float* __restrict__ bn,
                                 const float* __restrict__ Ws, const float* __restrict__ bs,
                                 const float* __restrict__ Wu, const float* __restrict__ bu,
                                 float* __restrict__ WnE, float* __restrict__ WsE,
                                 float* __restrict__ bE) {
  int i = blockIdx.x;   // output row 0..127
  int k = threadIdx.x;  // output col 0..127
  float an = 0.0f, as = 0.0f;
  for (int j = 0; j < 128; ++j) {
    float w1 = Wu[i * 256 + j];
    float w2 = Wu[i * 256 + 128 + j];
    an = fmaf(w1, Wn[j * 128 + k], an);
    as = fmaf(w2, Ws[j * 128 + k], as);
  }
  WnE[i * 128 + k] = an;
  WsE[i * 128 + k] = as;
  if (k == 0) {
    float b = bu[i];
    for (int j = 0; j < 128; ++j)
      b += Wu[i * 256 + j] * bn[j] + Wu[i * 256 + 128 + j] * bs[j];
    bE[i] = b;
  }
}

// ---------------- edge scatter (segment-sum + counts) ----------------
__global__ void scatter_sum_kernel(const float* __restrict__ feat, const int* __restrict__ ei,
                                   int E, float* __restrict__ agg, float* __restrict__ cnt) {
  int g = blockIdx.x * blockDim.x + threadIdx.x;
  int e = g >> 5;
  if (e >= E) return;
  int lane = threadIdx.x & 31;
  int src = ei[e];
  int dst = ei[E + e];
  const float4 v = *(const float4*)(feat + (long)src * 128 + lane * 4);
  float* a = agg + (long)dst * 128 + lane * 4;
  atomicAdd(a + 0, v.x);
  atomicAdd(a + 1, v.y);
  atomicAdd(a + 2, v.z);
  atomicAdd(a + 3, v.w);
  if (lane == 0) atomicAdd(cnt + dst, 1.0f);
}

// ---------------- fused multi-source GEMM via f32 WMMA ----------------
// out[m][n] = sum_s ( A_s[m][:] * 1/max(cnt_s[m],1) ) . W_s[n][:]  + bias[n]
template <int NSRC>
__global__ void gemm_wmma_kernel(const float* __restrict__ A0, const float* __restrict__ c0,
                                 const float* __restrict__ W0,
                                 const float* __restrict__ A1, const float* __restrict__ c1,
                                 const float* __restrict__ W1,
                                 const float* __restrict__ A2, const float* __restrict__ c2,
                                 const float* __restrict__ W2,
                                 const float* __restrict__ bias, float* __restrict__ out,
                                 int M) {
  extern __shared__ float lw[];  // NSRC * LDSP*128 floats
  const float* Aarr[3] = {A0, A1, A2};
  const float* carr[3] = {c0, c1, c2};
  const float* Warr[3] = {W0, W1, W2};

  int tid  = threadIdx.x;
  int wave = tid >> 5;
  int lane = tid & 31;

#if USE_TDM
  if (wave == 0) {
    unsigned base = __builtin_amdgcn_groupstaticsize();  // dynamic LDS starts after static (=0)
#pragma unroll
    for (int s = 0; s < NSRC; ++s)
      tdm_load_w128(Warr[s], base + (unsigned)s * (LDSP * 128u * 4u));
    __builtin_amdgcn_s_wait_tensorcnt(0);
  }
#else
#pragma unroll
  for (int s = 0; s < NSRC; ++s) {
    float* dst = lw + s * (LDSP * 128);
    for (int idx = tid; idx < 128 * 128; idx += 256) {
      int r = idx >> 7, c = idx & 127;
      dst[r * LDSP + c] = Warr[s][idx];
    }
  }
#endif
  __syncthreads();

  int col  = lane & 15;  // A-row / B-col within tile
  int half = lane >> 4;  // K-half select

  int tiles = M >> 4;  // M is a multiple of 16 here
  for (int tile = blockIdx.x * 8 + wave; tile < tiles; tile += gridDim.x * 8) {
    int m0   = tile << 4;
    int rowA = m0 + col;

    float rs[NSRC];
    const float* ab[NSRC];
#pragma unroll
    for (int s = 0; s < NSRC; ++s) {
      rs[s] = carr[s] ? 1.0f / fmaxf(carr[s][rowA], 1.0f) : 1.0f;
      ab[s] = Aarr[s] + (long)rowA * 128 + half * 2;
    }

    v8f acc[8] = {};

#pragma unroll 2
    for (int k0 = 0; k0 < 128; k0 += 4) {
      // batch all A-frags (global) and B-frags (LDS) first, then all WMMAs
      v2f a[NSRC];
#pragma unroll
      for (int s = 0; s < NSRC; ++s) {
        a[s] = *(const v2f*)(ab[s] + k0);
        a[s] *= rs[s];
      }
      v2f b[NSRC][8];
      const float* lb = lw + col * LDSP + k0 + half * 2;
#pragma unroll
      for (int s = 0; s < NSRC; ++s)
#pragma unroll
        for (int nt = 0; nt < 8; ++nt)
          b[s][nt] = *(const v2f*)(lb + s * (LDSP * 128) + nt * 16 * LDSP);
      // source-outer / n-tile-inner: consecutive WMMAs hit distinct accumulators
#pragma unroll
      for (int s = 0; s < NSRC; ++s)
#pragma unroll
        for (int nt = 0; nt < 8; ++nt)
          acc[nt] = __builtin_amdgcn_wmma_f32_16x16x4_f32(false, a[s], false, b[s][nt],
                                                          (short)0, acc[nt], false, false);
    }

    // C/D layout: VGPR r -> row m0 + r + half*8 ; col = nt*16 + (lane&15)
#pragma unroll
    for (int nt = 0; nt < 8; ++nt) {
      int cg   = nt * 16 + col;
      float bv = bias[cg];
      v8f cc   = acc[nt];
      float* ob = out + (long)(m0 + half * 8) * 128 + cg;
#pragma unroll
      for (int r = 0; r < 8; ++r) ob[(long)r * 128] = cc[r] + bv;
    }
  }
}

// ---------------- final link-prediction dot ----------------
__global__ void edge_dot_kernel(const float* __restrict__ h, const int* __restrict__ eli, int E,
                                float* __restrict__ out) {
  int g = blockIdx.x * blockDim.x + threadIdx.x;
  int e = g >> 5;
  if (e >= E) return;
  int lane = threadIdx.x & 31;
  int i0 = eli[e], i1 = eli[E + e];
  float4 a = *(const float4*)(h + (long)i0 * 128 + lane * 4);
  float4 b = *(const float4*)(h + (long)i1 * 128 + lane * 4);
  float d = a.x * b.x + a.y * b.y + a.z * b.z + a.w * b.w;
#pragma unroll
  for (int off = 16; off > 0; off >>= 1) d += __shfl_xor(d, off);
  if (lane == 0) out[e] = d;
}

// ---------------- host orchestration ----------------
extern "C" void kernel_launch(void* const* d_in, const int* in_sizes, int n_in,
                              void* d_out, int out_size, void* d_ws, size_t ws_size,
                              hipStream_t stream) {
  (void)in_sizes; (void)n_in; (void)out_size; (void)ws_size;
  const int N1 = 100000, N2 = 100000, E = 500000;

  const float* x1  = (const float*)d_in[0];
  const float* x2  = (const float*)d_in[1];
  const int*   ei1 = (const int*)d_in[2];
  const int*   ei2 = (const int*)d_in[3];
  const int*   ei3 = (const int*)d_in[4];
  const int*   eli = (const int*)d_in[5];
  auto P = [&](int layer, int mt, int which) -> const float* {
    return (const float*)d_in[6 + (layer * 3 + mt) * 6 + which];
  };

  float* ws = (float*)d_ws;
  size_t off = 0;
  auto alloc = [&](size_t n) { float* p = ws + off; off += n; return p; };
  float* A1 = alloc((size_t)N1 * 128);
  float* A2 = alloc((size_t)N2 * 128);
  float* G1 = alloc((size_t)N1 * 128);  // agg m1 (dst n1)
  float* G3 = alloc((size_t)N1 * 128);  // agg m3 (dst n1)
  float* G2 = alloc((size_t)N2 * 128);  // agg m2 (dst n2)
  float* O1 = alloc((size_t)N1 * 128);
  float* O2 = alloc((size_t)N2 * 128);
  float* C1 = alloc(N1);
  float* C3 = alloc(N1);
  float* C2 = alloc(N2);
  float *WnE[6], *WsE[6], *bE[6];
  for (int s = 0; s < 6; ++s) { WnE[s] = alloc(16384); WsE[s] = alloc(16384); bE[s] = alloc(128); }
  float *Wsum[2], *bsum[2];
  for (int l = 0; l < 2; ++l) { Wsum[l] = alloc(16384); bsum[l] = alloc(128); }

  // 1) fold weights (tiny)
  for (int s = 0; s < 6; ++s) {
    int l = s / 3, mt = s % 3;
    build_eff_kernel<<<128, 128, 0, stream>>>(P(l, mt, 0), P(l, mt, 1), P(l, mt, 2),
                                              P(l, mt, 3), P(l, mt, 4), P(l, mt, 5),
                                              WnE[s], WsE[s], bE[s]);
  }
  for (int l = 0; l < 2; ++l) {
    add2_kernel<<<(16384 + 255) / 256, 256, 0, stream>>>(WsE[l * 3 + 0], WsE[l * 3 + 2],
                                                         Wsum[l], 16384);
    add2_kernel<<<1, 128, 0, stream>>>(bE[l * 3 + 0], bE[l * 3 + 2], bsum[l], 128);
  }

  // 2) layer-1 activations
  {
    long n = (long)N1 * 128;
    leaky_kernel<<<(unsigned)((n + 255) / 256), 256, 0, stream>>>(x1, A1, n);
    n = (long)N2 * 128;
    leaky_kernel<<<(unsigned)((n + 255) / 256), 256, 0, stream>>>(x2, A2, n);
  }

  const unsigned edgeBlocks = (unsigned)(((long)E * 32 + 255) / 256);
  const int tiles1  = N1 / 16;
  const int tiles2  = N2 / 16;
  const int blocks1 = (tiles1 + 7) / 8 < 256 ? (tiles1 + 7) / 8 : 256;
  const int blocks2 = (tiles2 + 7) / 8 < 256 ? (tiles2 + 7) / 8 : 256;
  const size_t smem3 = 3u * LDSP * 128 * sizeof(float);
  const size_t smem2 = 2u * LDSP * 128 * sizeof(float);

  for (int l = 0; l < 2; ++l) {
    hipMemsetAsync(G1, 0, (size_t)(2 * N1 + N2) * 128 * sizeof(float), stream);
    hipMemsetAsync(C1, 0, (size_t)(2 * N1 + N2) * sizeof(float), stream);

    scatter_sum_kernel<<<edgeBlocks, 256, 0, stream>>>(A1, ei1, E, G1, C1);  // m1: a1 -> n1
    scatter_sum_kernel<<<edgeBlocks, 256, 0, stream>>>(A2, ei3, E, G3, C3);  // m3: a2 -> n1
    scatter_sum_kernel<<<edgeBlocks, 256, 0, stream>>>(A1, ei2, E, G2, C2);  // m2: a1 -> n2

    // out1 = mean1@WnE(m1)^T + mean3@WnE(m3)^T + a1@(WsE(m1)+WsE(m3))^T + bsum
    gemm_wmma_kernel<3><<<blocks1, 256, smem3, stream>>>(
        G1, C1, WnE[l * 3 + 0],
        G3, C3, WnE[l * 3 + 2],
        A1, nullptr, Wsum[l],
        bsum[l], O1, N1);
    // out2 = mean2@WnE(m2)^T + a2@WsE(m2)^T + bE(m2)
    gemm_wmma_kernel<2><<<blocks2, 256, smem2, stream>>>(
        G2, C2, WnE[l * 3 + 1],
        A2, nullptr, WsE[l * 3 + 1],
        nullptr, nullptr, nullptr,
        bE[l * 3 + 1], O2, N2);

    if (l == 0) {
      long n = (long)N1 * 128;
      leaky_kernel<<<(unsigned)((n + 255) / 256), 256, 0, stream>>>(O1, A1, n);
      n = (long)N2 * 128;
      leaky_kernel<<<(unsigned)((n + 255) / 256), 256, 0, stream>>>(O2, A2, n);
    }
  }

  // 3) link prediction: dot(h1[first], h1[second])
  edge_dot_kernel<<<edgeBlocks, 256, 0, stream>>>(O1, eli, E, (float*)d_out);
}